// MyGINConv_31456340476230
// MI455X (gfx1250) — compile-verified
//
#include <hip/hip_runtime.h>
#include <hip/hip_bf16.h>

typedef __attribute__((ext_vector_type(2))) float        v2f;
typedef __attribute__((ext_vector_type(8))) float        v8f;
typedef __attribute__((ext_vector_type(4))) unsigned int v4u;
typedef __attribute__((ext_vector_type(8))) int          v8i;
typedef __attribute__((ext_vector_type(4))) int          v4i;

#define N_NODES 100000
#define N_EDGES 640000
#define C_DIM   128
#define E_DIM   64
#define H_DIM   256      // 2*C
#define EB      64       // edges per block in edge kernel

// Padded LDS row strides (pad inserted by TDM pad_enable: +1 DWORD per row)
#define WE_STRIDE 129    // 128 + 1
#define EA_STRIDE 65     // 64 + 1

// ---------------------------------------------------------------------------
// Tensor Data Mover: 2D tile load Global -> LDS (clang-23 6-arg builtin).
// D# bitfields per CDNA5 ISA ch.8: group0 = {count|flags, lds_addr,
// global_addr[31:0], global_addr[56:32]|type=2}; group1 packs data_size,
// pad cfg, tensor dims, tile dims, stride0. Groups 2/3 zero (2D tensor).
// padctl = data_size(2=4B)<<16 | pad_enable<<20 | pad_interval<<22.
// ---------------------------------------------------------------------------
__device__ __forceinline__ void tdm_load_2d(unsigned lds_off, const void* gptr,
                                            unsigned tdim0, unsigned tdim1,
                                            unsigned tile0, unsigned tile1,
                                            unsigned stride0, unsigned padctl)
{
    const unsigned long long ga = (unsigned long long)(uintptr_t)gptr;
    v4u g0;
    g0.x = 1u;                                                  // count=1 (valid user D#)
    g0.y = lds_off;                                             // LDS byte address
    g0.z = (unsigned)ga;                                        // global addr [31:0]
    g0.w = ((unsigned)(ga >> 32) & 0x01FFFFFFu) | (2u << 30);   // addr[56:32] | type=2
    v8i g1;
    g1[0] = (int)padctl;                                        // mask=0,data_size,pad cfg
    g1[1] = (int)((tdim0 & 0xFFFFu) << 16);                     // tensor_dim0 lo16
    g1[2] = (int)(((tdim0 >> 16) & 0xFFFFu) | ((tdim1 & 0xFFFFu) << 16));
    g1[3] = (int)(((tdim1 >> 16) & 0xFFFFu) | ((tile0 & 0xFFFFu) << 16));
    g1[4] = (int)(tile1 & 0xFFFFu);                             // tile_dim1 | tile_dim2=0
    g1[5] = (int)stride0;                                       // tensor_dim0_stride lo32
    g1[6] = 0;
    g1[7] = 0;
    v4i z4 = {0, 0, 0, 0};
    v8i z8 = {0, 0, 0, 0, 0, 0, 0, 0};
    __builtin_amdgcn_tensor_load_to_lds(g0, g1, z4, z4, z8, 0);
}

// ---------------------------------------------------------------------------
// Kernel 1: edge projection GEMM + gather + ReLU + scatter-add
// Block: 256 threads (8 waves), 64 edges x 128 cols.
// TDM stages We (64x128) and the edge_attr tile (64x64) into LDS with
// +1-DWORD row padding; 8 waves run 4 x (16x16, K=64) WMMA tiles each.
// ---------------------------------------------------------------------------
__global__ void __launch_bounds__(256)
gin_edge_kernel(const float* __restrict__ edge_attr,   // [E, 64]
                const int*   __restrict__ edge_index,  // [2, E] row0=dst row1=src
                const float* __restrict__ x,           // [N, 128]
                const float* __restrict__ We,          // [64, 128]
                const float* __restrict__ be,          // [128]
                float*       __restrict__ accum)       // [N, 128]
{
    __shared__ float sWe[E_DIM * WE_STRIDE];   // 33024 B
    __shared__ float sEA[EB * EA_STRIDE];      // 16640 B

    const int wave  = threadIdx.x >> 5;      // 0..7 -> column tile
    const int lane  = threadIdx.x & 31;
    const int half  = lane >> 4;
    const int l16   = lane & 15;
    const int ebase = blockIdx.x * EB;
    const int n     = wave * 16 + l16;       // output column 0..127

    if (threadIdx.x < 32) {                  // wave 0 drives the TDM
        // We: 64x128 f32, row stride 128; pad every 128 DWORDs (code 6)
        tdm_load_2d((unsigned)(uintptr_t)&sWe[0], We,
                    C_DIM, E_DIM, C_DIM, E_DIM, C_DIM,
                    (2u << 16) | (1u << 20) | (6u << 22));
        // edge_attr tile: 64x64 f32, row stride 64; pad every 64 DWORDs (code 5)
        tdm_load_2d((unsigned)(uintptr_t)&sEA[0], edge_attr + (long)ebase * E_DIM,
                    E_DIM, EB, E_DIM, EB, E_DIM,
                    (2u << 16) | (1u << 20) | (5u << 22));
        __builtin_amdgcn_s_wait_tensorcnt(0);
    }
    __syncthreads();

    const float bias = be[n];
#pragma unroll
    for (int eg = 0; eg < EB / 16; ++eg) {
        const float* arow = &sEA[(eg * 16 + l16) * EA_STRIDE];
        v8f acc = {};
#pragma unroll
        for (int kb = 0; kb < E_DIM; kb += 4) {
            const int ka = kb + half * 2;
            v2f a; a.x = arow[ka];                 a.y = arow[ka + 1];
            v2f b; b.x = sWe[ka * WE_STRIDE + n];  b.y = sWe[(ka + 1) * WE_STRIDE + n];
            acc = __builtin_amdgcn_wmma_f32_16x16x4_f32(
                false, a, false, b, (short)0, acc, false, false);
        }
#pragma unroll
        for (int i = 0; i < 8; ++i) {
            const int m = i + half * 8;                // C/D layout row
            const int e = ebase + eg * 16 + m;
            const int src = edge_index[N_EDGES + e];   // edge_index[1][e]
            const int dst = edge_index[e];             // edge_index[0][e]
            float msg = acc[i] + bias + x[(long)src * C_DIM + n];
            msg = msg > 0.0f ? msg : 0.0f;
            atomicAdd(&accum[(long)dst * C_DIM + n], msg);
        }
    }
}

// ---------------------------------------------------------------------------
// Kernel 2: node MLP
//   h  = (1+eps)*x + accum ; h1 = relu(LN(h@W1+b1)) ; out = h1@W2 + b2
// ---------------------------------------------------------------------------
__global__ void __launch_bounds__(256)
gin_node_kernel(const float* __restrict__ x,
                const float* __restrict__ accum,
                const float* __restrict__ W1,     // [128, 256]
                const float* __restrict__ b1,     // [256]
                const float* __restrict__ gamma,  // [256]
                const float* __restrict__ beta,   // [256]
                const float* __restrict__ W2,     // [256, 128]
                const float* __restrict__ b2,     // [128]
                const float* __restrict__ eps,    // [1]
                float*       __restrict__ out)    // [N, 128]
{
    __shared__ float hs[16][C_DIM];     // 8 KB
    __shared__ float h1s[16][H_DIM];    // 16 KB
    __shared__ float psum[16][16];
    __shared__ float psq[16][16];

    const int nb   = blockIdx.x * 16;
    const int tid  = threadIdx.x;
    const int wave = tid >> 5;
    const int lane = tid & 31;
    const int half = lane >> 4;
    const int l16  = lane & 15;
    const float ep = 1.0f + eps[0];

    for (int i = tid; i < 16 * C_DIM; i += 256) {
        const int r = i >> 7, c = i & (C_DIM - 1);
        const long g = (long)(nb + r) * C_DIM + c;
        hs[r][c] = ep * x[g] + accum[g];
    }
    __syncthreads();

    // GEMM1: 16x128 @ 128x256 -> h1s
#pragma unroll
    for (int t = 0; t < 2; ++t) {
        const int n = (wave + t * 8) * 16 + l16;
        v8f acc = {};
#pragma unroll
        for (int kb = 0; kb < C_DIM; kb += 4) {
            const int ka = kb + half * 2;
            v2f a; a.x = hs[l16][ka];              a.y = hs[l16][ka + 1];
            v2f b; b.x = W1[(long)ka * H_DIM + n]; b.y = W1[(long)(ka + 1) * H_DIM + n];
            acc = __builtin_amdgcn_wmma_f32_16x16x4_f32(
                false, a, false, b, (short)0, acc, false, false);
        }
        const float bias = b1[n];
#pragma unroll
        for (int i = 0; i < 8; ++i)
            h1s[i + half * 8][n] = acc[i] + bias;
    }
    __syncthreads();

    // LayerNorm over 256 + ReLU, 16 threads per row
    const int row = tid >> 4;
    const int wi  = tid & 15;
    float s = 0.0f, sq = 0.0f;
    for (int c = wi; c < H_DIM; c += 16) {
        const float v = h1s[row][c];
        s += v; sq += v * v;
    }
    psum[row][wi] = s;
    psq[row][wi]  = sq;
    __syncthreads();
    float mean = 0.0f, m2 = 0.0f;
#pragma unroll
    for (int j = 0; j < 16; ++j) { mean += psum[row][j]; m2 += psq[row][j]; }
    mean *= (1.0f / H_DIM);
    const float var  = m2 * (1.0f / H_DIM) - mean * mean;
    const float rstd = rsqrtf(var + 1e-5f);
    for (int c = wi; c < H_DIM; c += 16) {
        float v = (h1s[row][c] - mean) * rstd * gamma[c] + beta[c];
        h1s[row][c] = v > 0.0f ? v : 0.0f;
    }
    __syncthreads();

    // GEMM2: 16x256 @ 256x128 -> out
    {
        const int n = wave * 16 + l16;
        v8f acc = {};
#pragma unroll
        for (int kb = 0; kb < H_DIM; kb += 4) {
            const int ka = kb + half * 2;
            v2f a; a.x = h1s[l16][ka];             a.y = h1s[l16][ka + 1];
            v2f b; b.x = W2[(long)ka * C_DIM + n]; b.y = W2[(long)(ka + 1) * C_DIM + n];
            acc = __builtin_amdgcn_wmma_f32_16x16x4_f32(
                false, a, false, b, (short)0, acc, false, false);
        }
        const float bias = b2[n];
#pragma unroll
        for (int i = 0; i < 8; ++i)
            out[(long)(nb + i + half * 8) * C_DIM + n] = acc[i] + bias;
    }
}

// ---------------------------------------------------------------------------
// Host launcher
// Input order: x, edge_index, edge_attr, We, be, W1, b1, gamma, beta, W2, b2, eps
// ---------------------------------------------------------------------------
extern "C" void kernel_launch(void* const* d_in, const int* in_sizes, int n_in,
                              void* d_out, int out_size, void* d_ws, size_t ws_size,
                              hipStream_t stream) {
    const float* x          = (const float*)d_in[0];
    const int*   edge_index = (const int*)  d_in[1];
    const float* edge_attr  = (const float*)d_in[2];
    const float* We         = (const float*)d_in[3];
    const float* be         = (const float*)d_in[4];
    const float* W1         = (const float*)d_in[5];
    const float* b1         = (const float*)d_in[6];
    const float* gamma      = (const float*)d_in[7];
    const float* beta       = (const float*)d_in[8];
    const float* W2         = (const float*)d_in[9];
    const float* b2         = (const float*)d_in[10];
    const float* eps        = (const float*)d_in[11];
    float* out   = (float*)d_out;
    float* accum = (float*)d_ws;   // [N_NODES, C_DIM]

    (void)hipMemsetAsync(accum, 0, (size_t)N_NODES * C_DIM * sizeof(float), stream);

    gin_edge_kernel<<<N_EDGES / EB, 256, 0, stream>>>(
        edge_attr, edge_index, x, We, be, accum);

    gin_node_kernel<<<N_NODES / 16, 256, 0, stream>>>(
        x, accum, W1, b1, gamma, beta, W2, b2, eps, out);
}